// BilateralFilter_51144470561476
// MI455X (gfx1250) — compile-verified
//
#include <hip/hip_runtime.h>
#include <hip/hip_bf16.h>
#include <stdint.h>

// Bilateral filter, d=9 (radius 4, circular support: 49 taps), sigmaColor=sigmaSpace=1.5,
// BORDER_REFLECT_101, CHW float32 image 3x1024x1024.
//
// Fused weight: space_w(dy,dx) * exp(-d1^2 * inv2s2) == exp2( -(dsq + d1^2) * inv2s2 * log2(e) )
// => exactly one v_exp_f32 per tap.

#define IMG_H 1024
#define IMG_W 1024
#define RAD   4
#define TILE  16
#define HALO  (TILE + 2 * RAD)   // 24
#define NELT  (3 * HALO * HALO)  // 1728

// -inv2s2 * log2(e) = -(0.5/2.25) * 1.4426950408889634
#define NEG_COEF (-0.32059889797532136f)

__device__ __forceinline__ int reflect101(int p, int n) {
    // radius (4) < n, single reflection suffices
    p = (p < 0) ? -p : p;
    p = (p >= n) ? (2 * n - 2 - p) : p;
    return p;
}

__global__ __launch_bounds__(256) void bilateral9_kernel(const float* __restrict__ img,
                                                         float* __restrict__ out) {
    __shared__ float tile[3][HALO][HALO];

    const int tid = threadIdx.y * TILE + threadIdx.x;   // 0..255
    const int x0 = blockIdx.x * TILE - RAD;
    const int y0 = blockIdx.y * TILE - RAD;

    // LDS byte offset of the tile: low 32 bits of the flat address of a __shared__
    // object are the LDS offset (LDS aperture lives in the high bits).
    const unsigned lds_base = (unsigned)(uintptr_t)(&tile[0][0][0]);
    const unsigned long long gbase = (unsigned long long)(uintptr_t)img;

    // ---- Async global -> LDS halo-tile load (gfx1250 async path, ASYNCcnt) ----
    // Per-lane source addresses implement BORDER_REFLECT_101 directly.
    for (int e = tid; e < NELT; e += 256) {
        int c   = e / (HALO * HALO);
        int rem = e - c * (HALO * HALO);
        int ty  = rem / HALO;
        int tx  = rem - ty * HALO;
        int gy  = reflect101(y0 + ty, IMG_H);
        int gx  = reflect101(x0 + tx, IMG_W);
        unsigned long long gaddr =
            gbase + 4ull * ((unsigned long long)(c * IMG_H + gy) * IMG_W + (unsigned)gx);
        unsigned laddr = lds_base + 4u * (unsigned)e;
        asm volatile("global_load_async_to_lds_b32 %0, %1, off"
                     :: "v"(laddr), "v"(gaddr)
                     : "memory");
    }
    asm volatile("s_wait_asynccnt 0x0" ::: "memory");
    __syncthreads();

    // ---- Compute: 49-tap circular bilateral stencil, one v_exp_f32 per tap ----
    const int lx = threadIdx.x + RAD;
    const int ly = threadIdx.y + RAD;

    const float c0 = tile[0][ly][lx];
    const float c1 = tile[1][ly][lx];
    const float c2 = tile[2][ly][lx];

    float n0 = 0.0f, n1 = 0.0f, n2 = 0.0f, den = 0.0f;

#pragma unroll
    for (int dy = -RAD; dy <= RAD; ++dy) {
#pragma unroll
        for (int dx = -RAD; dx <= RAD; ++dx) {
            if (dy * dy + dx * dx <= RAD * RAD) {   // compile-time pruned -> 49 live taps
                const float s0 = tile[0][ly + dy][lx + dx];
                const float s1 = tile[1][ly + dy][lx + dx];
                const float s2 = tile[2][ly + dy][lx + dx];
                const float d1 = fabsf(s0 - c0) + fabsf(s1 - c1) + fabsf(s2 - c2);
                // t = dsq + d1^2  (dsq is a literal after unrolling)
                const float t = __fmaf_rn(d1, d1, (float)(dy * dy + dx * dx));
                const float w = __builtin_amdgcn_exp2f(t * NEG_COEF);   // v_exp_f32
                n0 = __fmaf_rn(w, s0, n0);
                n1 = __fmaf_rn(w, s1, n1);
                n2 = __fmaf_rn(w, s2, n2);
                den += w;
            }
        }
    }

    // den >= 1 (center tap weight == 1): one IEEE divide + 3 muls instead of 3 divides.
    const float inv = 1.0f / den;

    const int gx = blockIdx.x * TILE + threadIdx.x;
    const int gy = blockIdx.y * TILE + threadIdx.y;
    const int p  = gy * IMG_W + gx;
    out[0 * IMG_H * IMG_W + p] = n0 * inv;
    out[1 * IMG_H * IMG_W + p] = n1 * inv;
    out[2 * IMG_H * IMG_W + p] = n2 * inv;
}

extern "C" void kernel_launch(void* const* d_in, const int* in_sizes, int n_in,
                              void* d_out, int out_size, void* d_ws, size_t ws_size,
                              hipStream_t stream) {
    (void)in_sizes; (void)n_in; (void)d_ws; (void)ws_size; (void)out_size;
    const float* img = (const float*)d_in[0];
    float* out = (float*)d_out;
    dim3 block(TILE, TILE);
    dim3 grid(IMG_W / TILE, IMG_H / TILE);
    bilateral9_kernel<<<grid, block, 0, stream>>>(img, out);
}